// ZoneClassifier_51994874085585
// MI455X (gfx1250) — compile-verified
//
#include <hip/hip_runtime.h>
#include <hip/hip_bf16.h>
#include <math.h>

// ---------------------------------------------------------------------------
// GAT(128 -> 8 heads x 32) + ELU + mean-pool + MLP(256->128->6) on gfx1250.
// Dominant cost: edge-wise gather/scatter (~1.1 GB through L2/HBM).
// GEMM x@W_gat runs on fp32 WMMA (v_wmma_f32_16x16x4_f32, wave32),
// 32x32 tile per wave (4 accumulators) for 4 WMMAs per load group.
// ---------------------------------------------------------------------------

typedef __attribute__((ext_vector_type(2))) float v2f;
typedef __attribute__((ext_vector_type(8))) float v8f;

#define HEADS 8
#define OUTC 32
#define HID 256          // HEADS*OUTC
#define FIN 128
#define NEG_SLOPE 0.2f

// ---------------- init: zero a float region -------------------------------
__global__ void k_zero(float* p, long long n) {
    long long i = (long long)blockIdx.x * blockDim.x + threadIdx.x;
    long long stride = (long long)gridDim.x * blockDim.x;
    for (; i < n; i += stride) p[i] = 0.0f;
}

// ---------------- GEMM: xh[N,256] = x[N,128] @ W[128,256] (fp32 WMMA) ------
// One wave computes a 32x32 output tile (2 M-frags x 2 N-frags).
// grid = (ceil(N/32), 2), block = 128 (4 waves); wave covers one of 8
// 32-column groups: ngrp = blockIdx.y*4 + wave.
__global__ void k_gemm_xh(const float* __restrict__ x,
                          const float* __restrict__ W,
                          float* __restrict__ xh, int N) {
    const int lane = threadIdx.x & 31;
    const int wave = threadIdx.x >> 5;               // 0..3
    const int m0 = blockIdx.x * 32;
    const int n0 = (blockIdx.y * 4 + wave) * 32;     // 0,32,...,224

    const int r0 = m0 + (lane & 15);
    const int c0 = n0 + (lane & 15);
    const int khalf = (lane >> 4) << 1;              // 0 or 2

    // clamp rows for loads so EXEC stays all-ones through the WMMA loop
    const int lr0 = (r0 < N) ? r0 : (N - 1);
    const int lr1 = (r0 + 16 < N) ? (r0 + 16) : (N - 1);
    const float* __restrict__ arow0 = x + (long long)lr0 * FIN + khalf;
    const float* __restrict__ arow1 = x + (long long)lr1 * FIN + khalf;
    const float* __restrict__ wcol0 = W + (long long)khalf * HID + c0;
    const float* __restrict__ wcol1 = wcol0 + 16;

    v8f c00 = {}, c01 = {}, c10 = {}, c11 = {};
#pragma unroll 2
    for (int k0 = 0; k0 < FIN; k0 += 4) {
        v2f a0 = *(const v2f*)(arow0 + k0);
        v2f a1 = *(const v2f*)(arow1 + k0);
        v2f b0, b1;
        long long wo = (long long)k0 * HID;
        b0[0] = wcol0[wo];       b0[1] = wcol0[wo + HID];
        b1[0] = wcol1[wo];       b1[1] = wcol1[wo + HID];
        c00 = __builtin_amdgcn_wmma_f32_16x16x4_f32(false, a0, false, b0,
                                                    (short)0, c00, false, false);
        c01 = __builtin_amdgcn_wmma_f32_16x16x4_f32(false, a0, false, b1,
                                                    (short)0, c01, false, false);
        c10 = __builtin_amdgcn_wmma_f32_16x16x4_f32(false, a1, false, b0,
                                                    (short)0, c10, false, false);
        c11 = __builtin_amdgcn_wmma_f32_16x16x4_f32(false, a1, false, b1,
                                                    (short)0, c11, false, false);
    }

    const int rb = m0 + ((lane >> 4) << 3);          // +0 or +8
    float* __restrict__ o0 = xh + (long long)rb * HID + c0;       // c00/c01 rows
    float* __restrict__ o1 = o0 + 16LL * HID;                     // c10/c11 rows

    if (m0 + 31 < N) {                               // fast path: no guards
#pragma unroll
        for (int r = 0; r < 8; ++r) {
            long long ro = (long long)r * HID;
            o0[ro]          = c00[r];
            o0[ro + 16]     = c01[r];
            o1[ro]          = c10[r];
            o1[ro + 16]     = c11[r];
        }
    } else {                                         // tail tile
#pragma unroll
        for (int r = 0; r < 8; ++r) {
            long long ro = (long long)r * HID;
            if (rb + r < N)      { o0[ro] = c00[r]; o0[ro + 16] = c01[r]; }
            if (rb + 16 + r < N) { o1[ro] = c10[r]; o1[ro + 16] = c11[r]; }
        }
    }
}

// ---------------- attention coefficients per node --------------------------
// a_src[n,h] = sum_c xh[n,h,c]*att_src[h,c]; same for dst. 1 thread per (n,h).
__global__ void k_attn_nh(const float* __restrict__ xh,
                          const float* __restrict__ att_src,
                          const float* __restrict__ att_dst,
                          float* __restrict__ a_src,
                          float* __restrict__ a_dst, int N) {
    int t = blockIdx.x * blockDim.x + threadIdx.x;
    if (t >= N * HEADS) return;
    int n = t >> 3, h = t & 7;
    const float* xv = xh + (long long)n * HID + h * OUTC;
    const float* as = att_src + h * OUTC;
    const float* ad = att_dst + h * OUTC;
    float ss = 0.f, sd = 0.f;
#pragma unroll 8
    for (int c = 0; c < OUTC; ++c) { float v = xv[c]; ss += v * as[c]; sd += v * ad[c]; }
    a_src[t] = ss;
    a_dst[t] = sd;
}

__device__ __forceinline__ float leaky(float a) {
    return a >= 0.f ? a : NEG_SLOPE * a;
}
__device__ __forceinline__ unsigned enc_f32(float f) {
    unsigned u = __float_as_uint(f);
    return (u & 0x80000000u) ? ~u : (u | 0x80000000u);
}
__device__ __forceinline__ float dec_f32(unsigned k) {
    return (k & 0x80000000u) ? __uint_as_float(k & 0x7FFFFFFFu)
                             : __uint_as_float(~k);
}

// ---------------- edge pass 1: segment max of leaky(alpha) -----------------
__global__ void k_edge_max(const int* __restrict__ ei,
                           const float* __restrict__ a_src,
                           const float* __restrict__ a_dst,
                           unsigned* __restrict__ mEnc, int E, int N) {
    int t = blockIdx.x * blockDim.x + threadIdx.x;
    int tot = (E + N) * HEADS;
    if (t >= tot) return;
    int e = t >> 3, h = t & 7;
    int s = (e < E) ? ei[e] : (e - E);
    int d = (e < E) ? ei[E + e] : (e - E);
    float al = leaky(a_src[s * HEADS + h] + a_dst[d * HEADS + h]);
    atomicMax(&mEnc[d * HEADS + h], enc_f32(al));
}

// ---------------- edge pass 2: denom = segment sum of exp(alpha - m) -------
__global__ void k_edge_denom(const int* __restrict__ ei,
                             const float* __restrict__ a_src,
                             const float* __restrict__ a_dst,
                             const unsigned* __restrict__ mEnc,
                             float* __restrict__ denom, int E, int N) {
    int t = blockIdx.x * blockDim.x + threadIdx.x;
    int tot = (E + N) * HEADS;
    if (t >= tot) return;
    int e = t >> 3, h = t & 7;
    int s = (e < E) ? ei[e] : (e - E);
    int d = (e < E) ? ei[E + e] : (e - E);
    int di = d * HEADS + h;
    float al = leaky(a_src[s * HEADS + h] + a_dst[di]);
    atomicAdd(&denom[di], expf(al - dec_f32(mEnc[di])));
}

// ---------------- edge pass 3: weighted scatter-add ------------------------
// 32 lanes per edge; lane owns 8 contiguous channels (one head per 4 lanes).
__global__ void k_edge_agg(const int* __restrict__ ei,
                           const float* __restrict__ a_src,
                           const float* __restrict__ a_dst,
                           const unsigned* __restrict__ mEnc,
                           const float* __restrict__ denom,
                           const float* __restrict__ xh,
                           float* __restrict__ agg, int E, int N) {
    int t = blockIdx.x * blockDim.x + threadIdx.x;
    int e = t >> 5, lane = t & 31;
    if (e >= E + N) return;
    int s = (e < E) ? ei[e] : (e - E);
    int d = (e < E) ? ei[E + e] : (e - E);
    int h = lane >> 2;
    int di = d * HEADS + h;
    float al = leaky(a_src[s * HEADS + h] + a_dst[di]);
    float coef = expf(al - dec_f32(mEnc[di])) / denom[di];

    int cb = lane * 8;
    const float4* xv = (const float4*)(xh + (long long)s * HID + cb);
    float4 v0 = xv[0], v1 = xv[1];
    float* o = agg + (long long)d * HID + cb;
    atomicAdd(o + 0, v0.x * coef);
    atomicAdd(o + 1, v0.y * coef);
    atomicAdd(o + 2, v0.z * coef);
    atomicAdd(o + 3, v0.w * coef);
    atomicAdd(o + 4, v1.x * coef);
    atomicAdd(o + 5, v1.y * coef);
    atomicAdd(o + 6, v1.z * coef);
    atomicAdd(o + 7, v1.w * coef);
}

// ---------------- ELU + mean pool (block-local partial sums) ---------------
#define POOL_ROWS 64
__global__ void k_elu_pool(const float* __restrict__ agg,
                           const float* __restrict__ b_gat,
                           float* __restrict__ pooled, int N) {
    int c = threadIdx.x;                 // 256 threads = channels
    int r0 = blockIdx.x * POOL_ROWS;
    float b = b_gat[c];
    float s = 0.f;
    for (int i = 0; i < POOL_ROWS; ++i) {
        int n = r0 + i;
        if (n < N) {
            float v = agg[(long long)n * HID + c] + b;
            s += (v > 0.f) ? v : (expf(v) - 1.f);
        }
    }
    atomicAdd(&pooled[c], s);
}

// ---------------- head MLP: 256 -> 128 (relu) -> 6 -------------------------
__global__ void k_head_mlp(const float* __restrict__ pooled,
                           const float* __restrict__ W1,
                           const float* __restrict__ b1,
                           const float* __restrict__ W2,
                           const float* __restrict__ b2,
                           float* __restrict__ out, int N) {
    __shared__ float g[HID];
    __shared__ float hid[HID / 2];
    int t = threadIdx.x;                 // 128 threads
    float inv = 1.0f / (float)N;
    g[t] = pooled[t] * inv;
    g[t + 128] = pooled[t + 128] * inv;
    __syncthreads();
    float s = b1[t];
    for (int c = 0; c < HID; ++c) s += g[c] * W1[c * 128 + t];
    hid[t] = (s > 0.f) ? s : 0.f;
    __syncthreads();
    if (t < 6) {
        float o = b2[t];
        for (int j = 0; j < 128; ++j) o += hid[j] * W2[j * 6 + t];
        out[t] = o;
    }
}

// ---------------------------------------------------------------------------
extern "C" void kernel_launch(void* const* d_in, const int* in_sizes, int n_in,
                              void* d_out, int out_size, void* d_ws, size_t ws_size,
                              hipStream_t stream) {
    const float* x       = (const float*)d_in[0];
    const int*   ei      = (const int*)d_in[1];   // edge_index [2,E] (int32)
    const float* W_gat   = (const float*)d_in[2];
    const float* att_src = (const float*)d_in[3];
    const float* att_dst = (const float*)d_in[4];
    const float* b_gat   = (const float*)d_in[5];
    const float* W1      = (const float*)d_in[6];
    const float* b1      = (const float*)d_in[7];
    const float* W2      = (const float*)d_in[8];
    const float* b2      = (const float*)d_in[9];
    float* out = (float*)d_out;

    const int N = in_sizes[0] / FIN;     // 50000
    const int E = in_sizes[1] / 2;       // 500000

    // workspace layout (floats)
    float* ws      = (float*)d_ws;
    float* xh      = ws;                                   // N*256
    float* a_src   = xh   + (long long)N * HID;            // N*8
    float* a_dst   = a_src + (long long)N * HEADS;         // N*8
    float* m_f     = a_dst + (long long)N * HEADS;         // N*8 (uint enc)
    float* denom   = m_f  + (long long)N * HEADS;          // N*8
    float* agg     = denom + (long long)N * HEADS;         // N*256
    float* pooled  = agg  + (long long)N * HID;            // 256
    unsigned* mEnc = (unsigned*)m_f;

    // zero [m_f .. pooled+256): contiguous region
    long long zn = (long long)N * HEADS * 2 + (long long)N * HID + HID;
    k_zero<<<2048, 256, 0, stream>>>(m_f, zn);

    // GEMM: 32x32 tile per wave; grid = (ceil(N/32), 2), 4 waves/block
    int mtiles = (N + 31) / 32;
    k_gemm_xh<<<dim3(mtiles, 2), 128, 0, stream>>>(x, W_gat, xh, N);

    // per-node attention logits
    int tnh = N * HEADS;
    k_attn_nh<<<(tnh + 255) / 256, 256, 0, stream>>>(xh, att_src, att_dst,
                                                     a_src, a_dst, N);

    // edge passes (E real edges + N self loops)
    int teh = (E + N) * HEADS;
    k_edge_max<<<(teh + 255) / 256, 256, 0, stream>>>(ei, a_src, a_dst, mEnc, E, N);
    k_edge_denom<<<(teh + 255) / 256, 256, 0, stream>>>(ei, a_src, a_dst, mEnc,
                                                        denom, E, N);
    long long tagg = (long long)(E + N) * 32;
    k_edge_agg<<<(int)((tagg + 255) / 256), 256, 0, stream>>>(ei, a_src, a_dst,
                                                              mEnc, denom, xh,
                                                              agg, E, N);

    // ELU + mean pool + head MLP
    k_elu_pool<<<(N + POOL_ROWS - 1) / POOL_ROWS, HID, 0, stream>>>(agg, b_gat,
                                                                    pooled, N);
    k_head_mlp<<<1, 128, 0, stream>>>(pooled, W1, b1, W2, b2, out, N);
}